// CustomLoss_34711925686778
// MI455X (gfx1250) — compile-verified
//
#include <hip/hip_runtime.h>
#include <math.h>

// ---------------------------------------------------------------------------
// Loss = sum of: 3x tube(recon, x) + tube(comple_out, labels_encoder)
//              + KL(mu, logvar) + CE(fusion_out, argmax(labels))
// Memory-bound reduction (~286 MB -> 1 float). Strategy:
//  * one pass over every byte with float4 loads
//  * per-wave row reductions for tube via V_WMMA_F32_16X16X4_F32 (ones-B
//    collapses the K dimension of a 16x4 partial-sum matrix -> row sums)
//  * deterministic 2-stage reduce: per-block partials in d_ws, single-block
//    tree reduce into d_out[0]. No float atomics, no persistent state.
// ---------------------------------------------------------------------------

typedef float v2f __attribute__((ext_vector_type(2)));
typedef float v8f __attribute__((ext_vector_type(8)));

#define B_ROWS   16384
#define D_DIM    512
#define C_DIM    100
#define ZN       (16384 * 128)       // mu/logvar element count

// d_ws layout (floats); every slot rewritten on every launch
#define TUBE_OFF 0                   // 4 pairs * 128 blocks = 512
#define KL_OFF   512                 // 256 blocks
#define CE_OFF   768                 // 256 blocks
#define WS_TOTAL 1024

// ---------------------------------------------------------------------------
// Tube partials. Grid (128, 4), block 256 (8 waves). Each wave owns 16 rows.
// Lane layout matches the 16x4 f32 WMMA A-matrix:
//   row m lives in lanes m and m+16; lane's v2f {.x,.y} are K-slots.
// Each of the 4 K-slots privately accumulates 128 of the 512 columns; the
// WMMA with an all-ones B matrix then reduces across K and across the lane
// halves, leaving per-row sums in the D accumulator (replicated over N).
// ---------------------------------------------------------------------------
__global__ __launch_bounds__(256)
void tube_partials(const float* __restrict__ p0a, const float* __restrict__ p0g,
                   const float* __restrict__ p1a, const float* __restrict__ p1g,
                   const float* __restrict__ p2a, const float* __restrict__ p2g,
                   const float* __restrict__ p3a, const float* __restrict__ p3g,
                   float* __restrict__ ws) {
  const int pair = blockIdx.y;
  const float* A;
  const float* G;
  if      (pair == 0) { A = p0a; G = p0g; }
  else if (pair == 1) { A = p1a; G = p1g; }
  else if (pair == 2) { A = p2a; G = p2g; }
  else                { A = p3a; G = p3g; }

  const int lane = threadIdx.x & 31;
  const int wave = threadIdx.x >> 5;
  const int m    = lane & 15;
  const int half = lane >> 4;
  const int rowBase = (blockIdx.x * 8 + wave) * 16;
  const size_t row  = (size_t)(rowBase + m);

  const float* arow = A + row * D_DIM;
  const float* grow = G + row * D_DIM;
  const int c0 = (half * 2) * 128;   // columns handled by this lane's .x slot
  const int c1 = c0 + 128;           // columns handled by this lane's .y slot

  v2f dAcc = {0.0f, 0.0f};
  v2f pAcc = {0.0f, 0.0f};
  v2f gAcc = {0.0f, 0.0f};

  #pragma unroll 4
  for (int i = 0; i < 128; i += 4) {
    float4 av = *reinterpret_cast<const float4*>(arow + c0 + i);
    float4 gv = *reinterpret_cast<const float4*>(grow + c0 + i);
    dAcc.x += av.x * gv.x + av.y * gv.y + av.z * gv.z + av.w * gv.w;
    pAcc.x += av.x * av.x + av.y * av.y + av.z * av.z + av.w * av.w;
    gAcc.x += gv.x * gv.x + gv.y * gv.y + gv.z * gv.z + gv.w * gv.w;

    float4 aw = *reinterpret_cast<const float4*>(arow + c1 + i);
    float4 gw = *reinterpret_cast<const float4*>(grow + c1 + i);
    dAcc.y += aw.x * gw.x + aw.y * gw.y + aw.z * gw.z + aw.w * gw.w;
    pAcc.y += aw.x * aw.x + aw.y * aw.y + aw.z * aw.z + aw.w * aw.w;
    gAcc.y += gw.x * gw.x + gw.y * gw.y + gw.z * gw.z + gw.w * gw.w;
  }

  // Cross-lane / cross-K reduction: D = A_partials x Ones + 0
  const v2f ones = {1.0f, 1.0f};
  v8f dD = {};
  v8f pD = {};
  v8f gD = {};
  dD = __builtin_amdgcn_wmma_f32_16x16x4_f32(false, dAcc, false, ones,
                                             (short)0, dD, false, false);
  pD = __builtin_amdgcn_wmma_f32_16x16x4_f32(false, pAcc, false, ones,
                                             (short)0, pD, false, false);
  gD = __builtin_amdgcn_wmma_f32_16x16x4_f32(false, gAcc, false, ones,
                                             (short)0, gD, false, false);

  // D layout: VGPR v, lanes 0-15 -> row rowBase+v ; lanes 16-31 -> rowBase+8+v
  // (values replicated across N, so every lane in a half holds all 8 rows).
  float s = 0.0f;
  #pragma unroll
  for (int v = 0; v < 8; ++v) {
    float dot = dD[v];
    float p2  = pD[v];
    float g2  = gD[v];
    float p = sqrtf(p2);
    float g = sqrtf(g2);
    float denom = p * g;
    float cosv  = (denom == 0.0f) ? 0.0f : dot / denom;
    float ss    = 1.0f - cosv * cosv;
    float sine  = (ss < 0.0f) ? 0.0f : sqrtf((ss <= 0.0f) ? 1e-8f : ss);
    float pc    = p * cosv;
    float r     = pc / ((g == 0.0f) ? (g + 1e-8f) : g);
    float base  = p * sine + fabsf(g - pc);
    float ds    = (r >= 1.0f) ? 0.5f * base
                : (r >= 0.0f) ? base
                              : 1.5f * fabsf(pc - g - p * sine);
    s += -logf(tanhf(1.0f / ds));
  }

  // total for this wave's 16 rows = half0 value + half1 value
  float tot = s + __shfl_xor(s, 16, 32);

  __shared__ float wsum[8];
  if (lane == 0) wsum[wave] = tot;
  __syncthreads();
  if (threadIdx.x == 0) {
    float t = 0.0f;
    #pragma unroll
    for (int i = 0; i < 8; ++i) t += wsum[i];
    // ALPHA = 1.0, mean over B rows
    ws[TUBE_OFF + pair * 128 + blockIdx.x] = t * (1.0f / (float)B_ROWS);
  }
}

// ---------------------------------------------------------------------------
// KL partials. Grid 256, block 256. Grid-stride over float4s.
// kl = -0.5*BETA*mean(1 + logvar - mu^2 - exp(logvar))
// ---------------------------------------------------------------------------
__global__ __launch_bounds__(256)
void kl_partials(const float* __restrict__ mu, const float* __restrict__ lv,
                 float* __restrict__ ws) {
  const int tid    = blockIdx.x * 256 + threadIdx.x;
  const int stride = gridDim.x * 256;
  const float4* m4 = reinterpret_cast<const float4*>(mu);
  const float4* l4 = reinterpret_cast<const float4*>(lv);
  const int n4 = ZN / 4;

  float acc = 0.0f;
  for (int i = tid; i < n4; i += stride) {
    float4 m = m4[i];
    float4 l = l4[i];
    acc += (1.0f + l.x - m.x * m.x - expf(l.x));
    acc += (1.0f + l.y - m.y * m.y - expf(l.y));
    acc += (1.0f + l.z - m.z * m.z - expf(l.z));
    acc += (1.0f + l.w - m.w * m.w - expf(l.w));
  }
  #pragma unroll
  for (int off = 16; off > 0; off >>= 1) acc += __shfl_xor(acc, off, 32);

  __shared__ float wsum[8];
  const int lane = threadIdx.x & 31;
  const int wave = threadIdx.x >> 5;
  if (lane == 0) wsum[wave] = acc;
  __syncthreads();
  if (threadIdx.x == 0) {
    float t = 0.0f;
    #pragma unroll
    for (int i = 0; i < 8; ++i) t += wsum[i];
    ws[blockIdx.x] = t * (-0.5f * 50000000.0f / (float)ZN);
  }
}

// ---------------------------------------------------------------------------
// Cross-entropy partials. Grid 256, block 256 (8 waves); one wave per row,
// grid-stride over the 16384 rows. C=100 -> <=4 columns per lane, cached in
// registers so logits are read once.
// ce_row = logsumexp(logits) - logits[argmax(labels)]
// ---------------------------------------------------------------------------
__global__ __launch_bounds__(256)
void ce_partials(const float* __restrict__ logits,
                 const float* __restrict__ labels,
                 float* __restrict__ ws) {
  const int lane   = threadIdx.x & 31;
  const int wave   = threadIdx.x >> 5;
  const int waveId = blockIdx.x * 8 + wave;
  const int nWaves = gridDim.x * 8;

  float acc = 0.0f;
  for (int row = waveId; row < B_ROWS; row += nWaves) {
    const float* lrow = logits + (size_t)row * C_DIM;
    const float* brow = labels + (size_t)row * C_DIM;

    float cache[4];
    int   nc = 0;
    float mymax = -INFINITY;
    float bv = -INFINITY;
    int   bi = 0x7fffffff;
    for (int c = lane; c < C_DIM; c += 32) {
      float l = lrow[c];
      cache[nc++] = l;
      mymax = fmaxf(mymax, l);
      float b = brow[c];
      if (b > bv) { bv = b; bi = c; }   // scanning ascending c keeps first-max
    }
    #pragma unroll
    for (int off = 16; off > 0; off >>= 1) {
      mymax = fmaxf(mymax, __shfl_xor(mymax, off, 32));
      float ov = __shfl_xor(bv, off, 32);
      int   oi = __shfl_xor(bi, off, 32);
      if (ov > bv || (ov == bv && oi < bi)) { bv = ov; bi = oi; }
    }
    float se = 0.0f;
    int   k = 0;
    for (int c = lane; c < C_DIM; c += 32) se += expf(cache[k++] - mymax);
    #pragma unroll
    for (int off = 16; off > 0; off >>= 1) se += __shfl_xor(se, off, 32);

    if (lane == 0) acc += (mymax + logf(se)) - lrow[bi];
  }

  __shared__ float wsum[8];
  if (lane == 0) wsum[wave] = acc;
  __syncthreads();
  if (threadIdx.x == 0) {
    float t = 0.0f;
    #pragma unroll
    for (int i = 0; i < 8; ++i) t += wsum[i];
    ws[blockIdx.x] = t * (1.0f / (float)B_ROWS);
  }
}

// ---------------------------------------------------------------------------
// Final deterministic reduce of the 1024 partials -> out[0].
// ---------------------------------------------------------------------------
__global__ __launch_bounds__(256)
void final_reduce(const float* __restrict__ ws, float* __restrict__ out) {
  __shared__ float s[256];
  float a = ws[threadIdx.x] + ws[threadIdx.x + 256] +
            ws[threadIdx.x + 512] + ws[threadIdx.x + 768];
  s[threadIdx.x] = a;
  __syncthreads();
  for (int off = 128; off > 0; off >>= 1) {
    if (threadIdx.x < off) s[threadIdx.x] += s[threadIdx.x + off];
    __syncthreads();
  }
  if (threadIdx.x == 0) out[0] = s[0];
}

// ---------------------------------------------------------------------------
extern "C" void kernel_launch(void* const* d_in, const int* in_sizes, int n_in,
                              void* d_out, int out_size, void* d_ws, size_t ws_size,
                              hipStream_t stream) {
  const float* fusion = (const float*)d_in[0];
  const float* comple = (const float*)d_in[1];
  const float* labels = (const float*)d_in[2];
  const float* labenc = (const float*)d_in[3];
  const float* xA     = (const float*)d_in[4];
  const float* xAr    = (const float*)d_in[5];
  const float* xB     = (const float*)d_in[6];
  const float* xBr    = (const float*)d_in[7];
  const float* xC     = (const float*)d_in[8];
  const float* xCr    = (const float*)d_in[9];
  const float* mu     = (const float*)d_in[10];
  const float* lv     = (const float*)d_in[11];

  float* out = (float*)d_out;
  float* ws  = (float*)d_ws;   // WS_TOTAL floats, fully rewritten each call

  tube_partials<<<dim3(128, 4), 256, 0, stream>>>(xAr, xA, xBr, xB, xCr, xC,
                                                  comple, labenc, ws);
  kl_partials<<<256, 256, 0, stream>>>(mu, lv, ws + KL_OFF);
  ce_partials<<<256, 256, 0, stream>>>(fusion, labels, ws + CE_OFF);
  final_reduce<<<1, 256, 0, stream>>>(ws, out);
}